// EnhancedMSTSN_71571335020990
// MI455X (gfx1250) — compile-verified
//
#include <hip/hip_runtime.h>
#include <hip/hip_bf16.h>
#include <math.h>

// ---------------------------------------------------------------------------
// EnhancedMSTSN forward for MI455X (gfx1250).
// Model is tiny (~1.5 GFLOP, ~9MB traffic) -> latency bound. Strategy:
// 5 fused kernels, LDS-resident activations, fp32 WMMA (v_wmma_f32_16x16x4_f32)
// for every dense matmul, and GLOBAL_LOAD_ASYNC_TO_LDS_B128 (ASYNCcnt) for all
// bulk global->LDS staging (no VGPR round-trip).
// ---------------------------------------------------------------------------

typedef __attribute__((ext_vector_type(2))) float v2f;
typedef __attribute__((ext_vector_type(8))) float v8f;

// ---- Async global->LDS copy (CDNA5 TDM-adjacent async path, ASYNCcnt).
// LDS destination VGPR = low 32 bits of the generic shared-memory address
// (ISA: LDS aperture generic addr has the LDS byte offset in addr[31:0]).
__device__ __forceinline__ void async_g2l_b128(void* lds_dst, const void* gsrc) {
  asm volatile("global_load_async_to_lds_b128 %0, %1, off"
               :: "v"((unsigned int)(uintptr_t)lds_dst), "v"(gsrc)
               : "memory");
}
__device__ __forceinline__ void async_wait0() {
  asm volatile("s_wait_asynccnt 0x0" ::: "memory");
}

// ---- WMMA helpers (16x16x4 f32). Layouts per CDNA5 ISA 7.12.2:
//  A 16x4 : lane m=lane&15; VGPR0 = K{0|2}, VGPR1 = K{1|3} (hi half-lanes +2)
//  B 4x16 : lane n=lane&15; VGPR0 = K{0|2}, VGPR1 = K{1|3}
//  C/D    : VGPR r -> row (r + 8*(lane>>4)), col lane&15
__device__ __forceinline__ v8f wmma_k4(v8f acc, const float* A, int lda,
                                       const float* B, int ldb) {
  const int lane = threadIdx.x & 31;
  const int half = lane >> 4, l = lane & 15;
  v2f a, b;
  a.x = A[l * lda + 2 * half + 0];
  a.y = A[l * lda + 2 * half + 1];
  b.x = B[(2 * half + 0) * ldb + l];
  b.y = B[(2 * half + 1) * ldb + l];
  return __builtin_amdgcn_wmma_f32_16x16x4_f32(false, a, false, b, (short)0,
                                               acc, false, false);
}

// Transposed-B variant: logical B(k,n) = Bt[n*ldb + k]  (for Q @ K^T)
__device__ __forceinline__ v8f wmma_k4_bt(v8f acc, const float* A, int lda,
                                          const float* Bt, int ldb) {
  const int lane = threadIdx.x & 31;
  const int half = lane >> 4, l = lane & 15;
  v2f a, b;
  a.x = A[l * lda + 2 * half + 0];
  a.y = A[l * lda + 2 * half + 1];
  b.x = Bt[l * ldb + 2 * half + 0];
  b.y = Bt[l * ldb + 2 * half + 1];
  return __builtin_amdgcn_wmma_f32_16x16x4_f32(false, a, false, b, (short)0,
                                               acc, false, false);
}

__device__ __forceinline__ float act_apply(float x, int act) {
  if (act == 1) return fmaxf(x, 0.f);
  if (act == 2) return 0.5f * x * (1.f + erff(x * 0.70710678118654752f)); // exact gelu
  return x;
}

// D[M,N] = act(A[M,K] @ B[K,N] + bias), row-major, operands in LDS.
// One 16x16 tile per wave iteration. `wave` must be wave-uniform (pass the
// readfirstlane'd wave id) so the tile loop is scalar and WMMA runs with
// EXEC all-ones. M,N multiples of 16; K multiple of 4; ntiles % nwaves == 0.
__device__ void gemm_lds(const float* A, const float* B, const float* bias,
                         float* D, int M, int N, int K, int act,
                         int wave, int nwaves) {
  const int ncols = N >> 4;
  const int ntiles = (M >> 4) * ncols;
  const int lane = threadIdx.x & 31;
  const int half = lane >> 4, l = lane & 15;
  for (int t = wave; t < ntiles; t += nwaves) {
    const int m0 = (t / ncols) << 4, n0 = (t % ncols) << 4;
    v8f c = {};
    for (int k = 0; k < K; k += 4)
      c = wmma_k4(c, A + m0 * K + k, K, B + k * N + n0, N);
    const float bb = bias ? bias[n0 + l] : 0.f;
#pragma unroll
    for (int r = 0; r < 8; ++r)
      D[(m0 + r + 8 * half) * N + n0 + l] = act_apply(c[r] + bb, act);
  }
}

// ---------------------------------------------------------------------------
// Kernel A: adaptive adjacency mask (one workgroup, 256 threads = 1 row each)
// ---------------------------------------------------------------------------
__global__ void k_adj(const float* __restrict__ emb, unsigned char* __restrict__ mask) {
  __shared__ float nrm[256 * 16];
  __shared__ float row2[256];
  __shared__ float thr_s;
  const int i = threadIdx.x;
  float v[16], ss = 0.f;
#pragma unroll
  for (int c = 0; c < 16; ++c) { v[c] = emb[i * 16 + c]; ss += v[c] * v[c]; }
  const float r = rsqrtf(fmaxf(ss, 1e-12f));
#pragma unroll
  for (int c = 0; c < 16; ++c) nrm[i * 16 + c] = v[c] * r;
  __syncthreads();
  float m1 = -1e30f, m2 = -1e30f;
  for (int j = 0; j < 256; ++j) {
    float d = 0.f;
#pragma unroll
    for (int c = 0; c < 16; ++c) d += nrm[i * 16 + c] * nrm[j * 16 + c];
    if (j == i) d = 0.f;
    if (d > m1) { m2 = m1; m1 = d; } else if (d > m2) m2 = d;
  }
  row2[i] = m2;
  __syncthreads();
  if (i == 0) {
    float t = row2[0];
    for (int j = 1; j < 256; ++j) t = fminf(t, row2[j]);
    thr_s = t;
  }
  __syncthreads();
  const float thr = thr_s;
  for (int j = 0; j < 256; ++j) {
    float d = 0.f;
#pragma unroll
    for (int c = 0; c < 16; ++c) d += nrm[i * 16 + c] * nrm[j * 16 + c];
    if (j == i) d = 0.f;
    if (d < thr) d = 0.f;
    mask[i * 256 + j] = (unsigned char)((d > 0.5f) || (j == i));
  }
}

// ---------------------------------------------------------------------------
// GAT softmax-aggregation: row i handled by one thread; NH heads, NC channels.
// ---------------------------------------------------------------------------
__device__ void gat_aggregate(const float* Hm, const float* sS, const float* sT,
                              const unsigned char* __restrict__ mr,
                              const float* __restrict__ bias, float* out,
                              int i, int NH, int NC, int relu) {
  const int HC = NH * NC;
  float mx[4] = {-1e30f, -1e30f, -1e30f, -1e30f};
  for (int j = 0; j < 256; ++j) {
    if (!mr[j]) continue;
    for (int h = 0; h < NH; ++h) {
      float e = sS[i * 4 + h] + sT[j * 4 + h];
      e = (e > 0.f) ? e : 0.2f * e;               // leaky_relu(0.2)
      mx[h] = fmaxf(mx[h], e);
    }
  }
  float acc[32];
  for (int c = 0; c < HC; ++c) acc[c] = 0.f;
  float den[4] = {0.f, 0.f, 0.f, 0.f};
  for (int j = 0; j < 256; ++j) {
    if (!mr[j]) continue;
    for (int h = 0; h < NH; ++h) {
      float e = sS[i * 4 + h] + sT[j * 4 + h];
      e = (e > 0.f) ? e : 0.2f * e;
      const float w = __expf(e - mx[h]);
      den[h] += w;
      for (int c = 0; c < NC; ++c)
        acc[h * NC + c] += w * Hm[j * HC + h * NC + c];
    }
  }
  for (int h = 0; h < NH; ++h) {
    const float inv = 1.f / den[h];
    for (int c = 0; c < NC; ++c) {
      float v = acc[h * NC + c] * inv + bias[h * NC + c];
      out[i * HC + h * NC + c] = relu ? fmaxf(v, 0.f) : v;
    }
  }
}

// ---------------------------------------------------------------------------
// Kernel B: spatial processor (proj + GAT1 + relu + GAT2). 1 WG per (b,s).
// ---------------------------------------------------------------------------
__global__ void k_spatial(const float* __restrict__ x,
                          const float* __restrict__ projW, const float* __restrict__ projb,
                          const float* __restrict__ W1g, const float* __restrict__ a1s,
                          const float* __restrict__ a1d, const float* __restrict__ b1,
                          const float* __restrict__ W2g, const float* __restrict__ a2s,
                          const float* __restrict__ a2d, const float* __restrict__ b2,
                          const unsigned char* __restrict__ mask,
                          float* __restrict__ spatial_out) {
  extern __shared__ float sm[];
  float* H0 = sm;            // [256][16]  (proj out, later GAT1 output)
  float* H1 = H0 + 4096;     // [256][16]  (gemm1 out)
  float* H2 = H1 + 4096;     // [256][32]  (gemm2 out)
  float* sS = H2 + 8192;     // [256][4]
  float* sT = sS + 1024;     // [256][4]
  float* wW1 = sT + 1024;    // 256
  float* wW2 = wW1 + 256;    // 512
  const int tid = threadIdx.x;
  const int wave = __builtin_amdgcn_readfirstlane(tid >> 5);  // uniform
  const int bs = blockIdx.x;
  // async-stage the GAT weight matrices into LDS
  if (tid < 64) async_g2l_b128(wW1 + tid * 4, W1g + tid * 4);
  if (tid < 128) async_g2l_b128(wW2 + tid * 4, W2g + tid * 4);
  {
    const float* xp = x + (size_t)(bs * 256 + tid) * 3;
    const float x0 = xp[0], x1 = xp[1], x2 = xp[2];
#pragma unroll
    for (int c = 0; c < 16; ++c)
      H0[tid * 16 + c] = x0 * projW[c] + x1 * projW[16 + c] + x2 * projW[32 + c] + projb[c];
  }
  async_wait0();
  __syncthreads();
  gemm_lds(H0, wW1, nullptr, H1, 256, 16, 16, 0, wave, 8);   // h = H0 @ W1
  __syncthreads();
  for (int h = 0; h < 4; ++h) {
    float s = 0.f, t = 0.f;
#pragma unroll
    for (int c = 0; c < 4; ++c) {
      const float hv = H1[tid * 16 + h * 4 + c];
      s += hv * a1s[h * 4 + c];
      t += hv * a1d[h * 4 + c];
    }
    sS[tid * 4 + h] = s; sT[tid * 4 + h] = t;
  }
  __syncthreads();
  gat_aggregate(H1, sS, sT, mask + tid * 256, b1, H0, tid, 4, 4, 1);
  __syncthreads();
  gemm_lds(H0, wW2, nullptr, H2, 256, 32, 16, 0, wave, 8);   // h = H0 @ W2
  __syncthreads();
  for (int h = 0; h < 4; ++h) {
    float s = 0.f, t = 0.f;
#pragma unroll
    for (int c = 0; c < 8; ++c) {
      const float hv = H2[tid * 32 + h * 8 + c];
      s += hv * a2s[h * 8 + c];
      t += hv * a2d[h * 8 + c];
    }
    sS[tid * 4 + h] = s; sT[tid * 4 + h] = t;
  }
  __syncthreads();
  gat_aggregate(H2, sS, sT, mask + tid * 256, b2,
                spatial_out + (size_t)bs * 256 * 32, tid, 4, 8, 0);
}

// ---------------------------------------------------------------------------
// Kernel C: spatial_feats[b,n,c] = mean_s spatial_out[b,s,n,c]
// ---------------------------------------------------------------------------
__global__ void k_smean(const float* __restrict__ sp, float* __restrict__ sf) {
  const int idx = blockIdx.x * blockDim.x + threadIdx.x;
  if (idx >= 8 * 256 * 32) return;
  const int c = idx & 31, n = (idx >> 5) & 255, b = idx >> 13;
  float a = 0.f;
  for (int s = 0; s < 32; ++s)
    a += sp[(((size_t)(b * 32 + s) * 256 + n) << 5) + c];
  sf[idx] = a * (1.f / 32.f);
}

// ---------------------------------------------------------------------------
// Kernel D: temporal transformer, one WG (128 thr / 4 waves) per sequence p.
// Emits the per-sequence mean over time directly into temporal_feats.
// ---------------------------------------------------------------------------
__global__ void k_temporal(const float* __restrict__ sp,
    const float* __restrict__ gWq, const float* __restrict__ gbq,
    const float* __restrict__ gWk, const float* __restrict__ gbk,
    const float* __restrict__ gWv, const float* __restrict__ gbv,
    const float* __restrict__ gWo, const float* __restrict__ gbo,
    const float* __restrict__ g1g, const float* __restrict__ g1b,
    const float* __restrict__ g2g, const float* __restrict__ g2b,
    const float* __restrict__ gW1, const float* __restrict__ gb1,
    const float* __restrict__ gW2, const float* __restrict__ gb2,
    float* __restrict__ tfeat) {
  extern __shared__ float sm[];
  float* X  = sm;            // [32][32]  (input, later Y2)
  float* Q  = X + 1024;
  float* Km = Q + 1024;
  float* V  = Km + 1024;
  float* SC = V + 1024;      // [2][32][32] scores -> alpha
  float* O  = SC + 2048;     // [32][32]
  float* AT = O + 1024;      // attn out
  float* Y1 = AT + 1024;
  float* HF = Y1 + 1024;     // [32][64]
  float* Z  = HF + 2048;
  float* Wq = Z + 1024;
  float* Wk = Wq + 1024;
  float* Wv = Wk + 1024;
  float* Wo = Wv + 1024;
  float* W1 = Wo + 1024;     // [32][64]
  float* W2 = W1 + 2048;     // [64][32]
  float* SB = W2 + 2048;     // biases / ln params (384)
  const int tid = threadIdx.x;
  const int wave = __builtin_amdgcn_readfirstlane(tid >> 5);  // uniform
  const int p = blockIdx.x;

  // async-stage activations + weights straight into LDS (ASYNCcnt path)
  for (int t = tid * 4; t < 1024; t += 512) {
    async_g2l_b128(X + t,  sp + (size_t)p * 1024 + t);
    async_g2l_b128(Wq + t, gWq + t);
    async_g2l_b128(Wk + t, gWk + t);
    async_g2l_b128(Wv + t, gWv + t);
    async_g2l_b128(Wo + t, gWo + t);
  }
  for (int t = tid * 4; t < 2048; t += 512) {
    async_g2l_b128(W1 + t, gW1 + t);
    async_g2l_b128(W2 + t, gW2 + t);
  }
  if (tid < 32) {
    SB[tid] = gbq[tid]; SB[32 + tid] = gbk[tid]; SB[64 + tid] = gbv[tid];
    SB[96 + tid] = gbo[tid];
    SB[128 + tid] = gb1[tid]; SB[160 + tid] = gb1[32 + tid];
    SB[192 + tid] = gb2[tid];
    SB[224 + tid] = g1g[tid]; SB[256 + tid] = g1b[tid];
    SB[288 + tid] = g2g[tid]; SB[320 + tid] = g2b[tid];
  }
  async_wait0();
  __syncthreads();
  gemm_lds(X, Wq, SB + 0,  Q,  32, 32, 32, 0, wave, 4);
  gemm_lds(X, Wk, SB + 32, Km, 32, 32, 32, 0, wave, 4);
  gemm_lds(X, Wv, SB + 64, V,  32, 32, 32, 0, wave, 4);
  __syncthreads();
  // scores: SC[h] = (Q_h @ K_h^T) / sqrt(16)
  {
    const int lane = threadIdx.x & 31, half = lane >> 4, l = lane & 15;
    for (int t = wave; t < 8; t += 4) {
      const int h = t >> 2, m0 = ((t >> 1) & 1) << 4, n0 = (t & 1) << 4;
      v8f c = {};
      for (int k = 0; k < 16; k += 4)
        c = wmma_k4_bt(c, Q + m0 * 32 + h * 16 + k, 32,
                          Km + n0 * 32 + h * 16 + k, 32);
#pragma unroll
      for (int r = 0; r < 8; ++r)
        SC[h * 1024 + (m0 + r + 8 * half) * 32 + n0 + l] = c[r] * 0.25f;
    }
  }
  __syncthreads();
  if (tid < 64) {                                   // softmax over keys
    float* row = SC + (tid >> 5) * 1024 + (tid & 31) * 32;
    float m = row[0];
    for (int j = 1; j < 32; ++j) m = fmaxf(m, row[j]);
    float s = 0.f;
    for (int j = 0; j < 32; ++j) { const float w = __expf(row[j] - m); row[j] = w; s += w; }
    const float inv = 1.f / s;
    for (int j = 0; j < 32; ++j) row[j] *= inv;
  }
  __syncthreads();
  // O_h = alpha_h @ V_h   (4 tiles, one per wave)
  {
    const int lane = threadIdx.x & 31, half = lane >> 4, l = lane & 15;
    const int h = wave >> 1, m0 = (wave & 1) << 4;
    v8f c = {};
    for (int k = 0; k < 32; k += 4)
      c = wmma_k4(c, SC + h * 1024 + m0 * 32 + k, 32, V + k * 32 + h * 16, 32);
#pragma unroll
    for (int r = 0; r < 8; ++r)
      O[(m0 + r + 8 * half) * 32 + h * 16 + l] = c[r];
  }
  __syncthreads();
  gemm_lds(O, Wo, SB + 96, AT, 32, 32, 32, 0, wave, 4);
  __syncthreads();
  if (tid < 32) {                                   // LN1(X + attn)
    float v[32]; float m = 0.f;
    for (int c = 0; c < 32; ++c) { v[c] = X[tid * 32 + c] + AT[tid * 32 + c]; m += v[c]; }
    m *= (1.f / 32.f);
    float var = 0.f;
    for (int c = 0; c < 32; ++c) { const float d = v[c] - m; var += d * d; }
    const float inv = rsqrtf(var * (1.f / 32.f) + 1e-3f);
    for (int c = 0; c < 32; ++c)
      Y1[tid * 32 + c] = (v[c] - m) * inv * SB[224 + c] + SB[256 + c];
  }
  __syncthreads();
  gemm_lds(Y1, W1, SB + 128, HF, 32, 64, 32, 2, wave, 4);   // gelu FFN up
  __syncthreads();
  gemm_lds(HF, W2, SB + 192, Z, 32, 32, 64, 0, wave, 4);    // FFN down
  __syncthreads();
  if (tid < 32) {                                   // LN2(Y1 + Z) -> X (reuse)
    float v[32]; float m = 0.f;
    for (int c = 0; c < 32; ++c) { v[c] = Y1[tid * 32 + c] + Z[tid * 32 + c]; m += v[c]; }
    m *= (1.f / 32.f);
    float var = 0.f;
    for (int c = 0; c < 32; ++c) { const float d = v[c] - m; var += d * d; }
    const float inv = rsqrtf(var * (1.f / 32.f) + 1e-3f);
    for (int c = 0; c < 32; ++c)
      X[tid * 32 + c] = (v[c] - m) * inv * SB[288 + c] + SB[320 + c];
  }
  __syncthreads();
  if (tid < 32) {                                   // mean over time
    float a = 0.f;
    for (int s = 0; s < 32; ++s) a += X[s * 32 + tid];
    tfeat[(size_t)p * 32 + tid] = a * (1.f / 32.f);
  }
}

// ---------------------------------------------------------------------------
// Kernel E: cross-attention fusion + regressor. 1 WG (256 thr) per batch b.
// K/V live in LDS (exploits CDNA5's 320KB WGP LDS).
// ---------------------------------------------------------------------------
__global__ void k_fuse(const float* __restrict__ sfg, const float* __restrict__ tfg,
    const float* __restrict__ gWq, const float* __restrict__ gbq,
    const float* __restrict__ gWk, const float* __restrict__ gbk,
    const float* __restrict__ gWv, const float* __restrict__ gbv,
    const float* __restrict__ gWo, const float* __restrict__ gbo,
    const float* __restrict__ rW1g, const float* __restrict__ rb1,
    const float* __restrict__ rW2, const float* __restrict__ rb2,
    float* __restrict__ out) {
  extern __shared__ float sm[];
  float* SF = sm;              // [256][32]; SF..TF region reused as O [256][64]
  float* TF = SF + 8192;
  float* Kf = SF + 16384;      // [256][64]; later reused as fused F [256][32]
  float* Vf = Kf + 16384;      // [256][64]
  float* Wq = Vf + 16384;      // [32][64]
  float* Wk = Wq + 2048;
  float* Wv = Wk + 2048;
  float* Wo = Wv + 2048;       // [64][32]
  float* W1 = Wo + 2048;       // rW1 [32][16]
  float* SB = W1 + 512;        // biases (256)
  const int tid = threadIdx.x;
  const int wave = __builtin_amdgcn_readfirstlane(tid >> 5);  // uniform
  const int b = blockIdx.x;

  // async-stage features + projection weights into LDS
  for (int t = tid * 4; t < 8192; t += 1024) {
    async_g2l_b128(SF + t, sfg + (size_t)b * 8192 + t);
    async_g2l_b128(TF + t, tfg + (size_t)b * 8192 + t);
  }
  for (int t = tid * 4; t < 2048; t += 1024) {
    async_g2l_b128(Wq + t, gWq + t);
    async_g2l_b128(Wk + t, gWk + t);
    async_g2l_b128(Wv + t, gWv + t);
    async_g2l_b128(Wo + t, gWo + t);
  }
  if (tid < 128) async_g2l_b128(W1 + tid * 4, rW1g + tid * 4);
  if (tid < 64) { SB[tid] = gbq[tid]; SB[64 + tid] = gbk[tid]; SB[128 + tid] = gbv[tid]; }
  if (tid < 32) SB[192 + tid] = gbo[tid];
  if (tid < 16) SB[224 + tid] = rb1[tid];
  if (tid == 0) SB[240] = rb2[0];
  async_wait0();
  __syncthreads();
  gemm_lds(TF, Wk, SB + 64,  Kf, 256, 64, 32, 0, wave, 8);  // K = TF @ Wk
  gemm_lds(TF, Wv, SB + 128, Vf, 256, 64, 32, 0, wave, 8);  // V = TF @ Wv
  __syncthreads();
  // q_i in registers (reads only own SF row)
  const int i = tid;
  float q[64];
#pragma unroll 4
  for (int hc = 0; hc < 64; ++hc) {
    float a = SB[hc];
    for (int d = 0; d < 32; ++d) a += SF[i * 32 + d] * Wq[d * 64 + hc];
    q[hc] = a;
  }
  __syncthreads();             // SF/TF reads complete -> region becomes O
  const float scale = 0.17677669529663689f;  // 1/sqrt(32)
  for (int h = 0; h < 2; ++h) {
    float m = -1e30f;
    for (int j = 0; j < 256; ++j) {
      float d = 0.f;
#pragma unroll
      for (int c = 0; c < 32; ++c) d += q[h * 32 + c] * Kf[j * 64 + h * 32 + c];
      m = fmaxf(m, d * scale);
    }
    float den = 0.f, acc[32];
#pragma unroll
    for (int c = 0; c < 32; ++c) acc[c] = 0.f;
    for (int j = 0; j < 256; ++j) {
      float d = 0.f;
#pragma unroll
      for (int c = 0; c < 32; ++c) d += q[h * 32 + c] * Kf[j * 64 + h * 32 + c];
      const float w = __expf(d * scale - m);
      den += w;
#pragma unroll
      for (int c = 0; c < 32; ++c) acc[c] += w * Vf[j * 64 + h * 32 + c];
    }
    const float inv = 1.f / den;
#pragma unroll
    for (int c = 0; c < 32; ++c) SF[i * 64 + h * 32 + c] = acc[c] * inv; // O row
  }
  __syncthreads();
  float* F = Kf;               // fused [256][32], reuse K region
  gemm_lds(SF, Wo, SB + 192, F, 256, 32, 64, 0, wave, 8);   // fused = O @ Wo
  __syncthreads();
  {                            // regressor: gelu(F @ rW1 + rb1) @ rW2 + rb2
    float res = SB[240];
#pragma unroll
    for (int k = 0; k < 16; ++k) {
      float hh = SB[224 + k];
      for (int c = 0; c < 32; ++c) hh += F[i * 32 + c] * W1[c * 16 + k];
      hh = 0.5f * hh * (1.f + erff(hh * 0.70710678118654752f));
      res += hh * rW2[k];
    }
    out[b * 256 + i] = res;
  }
}

// ---------------------------------------------------------------------------
// Host launcher. Input order = setup_inputs() dict order (nested insertion
// order): x, emb, proj_W, proj_b, gat1{W,a_src,a_dst,b}, gat2{...}, tWq, tbq,
// tWk, tbk, tWv, tbv, tWo, tbo, ln1_g, ln1_b, ln2_g, ln2_b, fW1, fb1, fW2,
// fb2, cWq, cbq, cWk, cbk, cWv, cbv, cWo, cbo, rW1, rb1, rW2, rb2.
// ---------------------------------------------------------------------------
extern "C" void kernel_launch(void* const* d_in, const int* in_sizes, int n_in,
                              void* d_out, int out_size, void* d_ws, size_t ws_size,
                              hipStream_t stream) {
  (void)in_sizes; (void)n_in; (void)out_size; (void)ws_size;
  const float* x     = (const float*)d_in[0];
  const float* emb   = (const float*)d_in[1];
  const float* projW = (const float*)d_in[2];
  const float* projb = (const float*)d_in[3];
  const float* g1W   = (const float*)d_in[4];
  const float* g1as  = (const float*)d_in[5];
  const float* g1ad  = (const float*)d_in[6];
  const float* g1b   = (const float*)d_in[7];
  const float* g2W   = (const float*)d_in[8];
  const float* g2as  = (const float*)d_in[9];
  const float* g2ad  = (const float*)d_in[10];
  const float* g2b   = (const float*)d_in[11];
  const float* tWq = (const float*)d_in[12]; const float* tbq = (const float*)d_in[13];
  const float* tWk = (const float*)d_in[14]; const float* tbk = (const float*)d_in[15];
  const float* tWv = (const float*)d_in[16]; const float* tbv = (const float*)d_in[17];
  const float* tWo = (const float*)d_in[18]; const float* tbo = (const float*)d_in[19];
  const float* ln1g = (const float*)d_in[20]; const float* ln1b = (const float*)d_in[21];
  const float* ln2g = (const float*)d_in[22]; const float* ln2b = (const float*)d_in[23];
  const float* fW1 = (const float*)d_in[24]; const float* fb1 = (const float*)d_in[25];
  const float* fW2 = (const float*)d_in[26]; const float* fb2 = (const float*)d_in[27];
  const float* cWq = (const float*)d_in[28]; const float* cbq = (const float*)d_in[29];
  const float* cWk = (const float*)d_in[30]; const float* cbk = (const float*)d_in[31];
  const float* cWv = (const float*)d_in[32]; const float* cbv = (const float*)d_in[33];
  const float* cWo = (const float*)d_in[34]; const float* cbo = (const float*)d_in[35];
  const float* rW1 = (const float*)d_in[36]; const float* rb1 = (const float*)d_in[37];
  const float* rW2 = (const float*)d_in[38]; const float* rb2 = (const float*)d_in[39];

  char* ws = (char*)d_ws;
  unsigned char* mask = (unsigned char*)ws;                 //   64 KB
  float* spatial = (float*)(ws + 65536);                    //    8 MB
  float* tfeat   = (float*)(ws + 65536 + 8388608);          //  256 KB
  float* sfeat   = (float*)(ws + 65536 + 8388608 + 262144); //  256 KB

  k_adj<<<1, 256, 0, stream>>>(emb, mask);

  const size_t smB = (4096 + 4096 + 8192 + 1024 + 1024 + 256 + 512) * sizeof(float);
  k_spatial<<<256, 256, smB, stream>>>(x, projW, projb, g1W, g1as, g1ad, g1b,
                                       g2W, g2as, g2ad, g2b, mask, spatial);

  k_smean<<<(8 * 256 * 32 + 255) / 256, 256, 0, stream>>>(spatial, sfeat);

  const size_t smD = (1024 * 8 + 2048 * 2 + 4096 * 2 + 2048 + 384) * sizeof(float);
  k_temporal<<<2048, 128, smD, stream>>>(spatial, tWq, tbq, tWk, tbk, tWv, tbv,
                                         tWo, tbo, ln1g, ln1b, ln2g, ln2b,
                                         fW1, fb1, fW2, fb2, tfeat);

  const size_t smE = (16384 * 3 + 2048 * 4 + 512 + 256) * sizeof(float);
  k_fuse<<<8, 256, smE, stream>>>(sfeat, tfeat, cWq, cbq, cWk, cbk, cWv, cbv,
                                  cWo, cbo, rW1, rb1, rW2, rb2, (float*)d_out);
}